// StringNumericEmbedding_91096256348658
// MI455X (gfx1250) — compile-verified
//
#include <hip/hip_runtime.h>

// StringNumericEmbedding for MI455X (gfx1250, wave32).
//
// out[n, :] = is_numeric[n] ? values[n] * W[:,0] + b : emb_table[embedding_idx[n], :]
//
// N = 262144, D = 512, V = 50258.
// Memory-bound: 512 MB NT writes + ~103 MB L2-resident table reads -> ~26 us floor
// at 23.3 TB/s. One row per wave32: uniform branch, scalar per-row metadata loads,
// 4x b128 loads / 4x b128 non-temporal stores per lane.

typedef float v4f __attribute__((ext_vector_type(4)));

#ifndef EMB_D
#define EMB_D 512
#endif
static constexpr int kD  = EMB_D;     // d_model
static constexpr int kD4 = kD / 4;    // float4 per row = 128
static constexpr int kBlock = 256;    // 8 wave32 per block -> 8 rows per block

__global__ __launch_bounds__(kBlock)
void string_numeric_embedding_kernel(
    const float* __restrict__ values,      // [N]
    const int*   __restrict__ is_numeric,  // [N] (0/1)
    const int*   __restrict__ emb_idx,     // [N]
    const float* __restrict__ emb_table,   // [V, D]
    const float* __restrict__ W,           // [D, 1] -> contiguous D floats
    const float* __restrict__ b,           // [D]
    float*       __restrict__ out,         // [N, D]
    int N)
{
    const int lane = threadIdx.x & 31;          // lane within wave32
    const int wave = threadIdx.x >> 5;          // wave within block (0..7)
    const int row  = blockIdx.x * (kBlock >> 5) + wave;
    if (row >= N) return;

    // row is uniform across the wave by construction; force it into an SGPR so
    // the per-row metadata loads become scalar s_load_b32 (1 fetch per wave,
    // not 32 replicated lane loads).
    const int row_u = __builtin_amdgcn_readfirstlane(row);

    v4f* __restrict__ outv = reinterpret_cast<v4f*>(out) + (size_t)row_u * kD4;

    const bool numeric = (is_numeric[row_u] != 0);   // wave-uniform branch

    if (numeric) {
        const float v = values[row_u];               // uniform scalar
        const v4f* __restrict__ Wv = reinterpret_cast<const v4f*>(W);
        const v4f* __restrict__ bv = reinterpret_cast<const v4f*>(b);
#pragma unroll
        for (int i = 0; i < kD4 / 32; ++i) {         // 4 float4 per lane
            const int c = lane + 32 * i;
            const v4f w  = Wv[c];                    // L1-resident (2 KB)
            const v4f bb = bv[c];
            const v4f o  = v * w + bb;               // v_fma_f32 x4
            // NT store: write-once 512 MB stream must not evict the
            // 103 MB embedding table from the 192 MB L2.
            __builtin_nontemporal_store(o, &outv[c]);
        }
    } else {
        const int idx = __builtin_amdgcn_readfirstlane(emb_idx[row_u]);
        const v4f* __restrict__ ev =
            reinterpret_cast<const v4f*>(emb_table + (size_t)idx * kD);
#pragma unroll
        for (int i = 0; i < kD4 / 32; ++i) {
            const int c = lane + 32 * i;
            const v4f e = ev[c];                     // regular-temporal: keep table in L2
            __builtin_nontemporal_store(e, &outv[c]);
        }
    }
}

extern "C" void kernel_launch(void* const* d_in, const int* in_sizes, int n_in,
                              void* d_out, int out_size, void* d_ws, size_t ws_size,
                              hipStream_t stream)
{
    // setup_inputs() order:
    //   0: values        float32 [N]
    //   1: is_numeric    bool->int [N]
    //   2: embedding_idx int32 [N]
    //   3: emb_table     float32 [V, D]
    //   4: W             float32 [D, 1]
    //   5: b             float32 [D]
    const float* values     = static_cast<const float*>(d_in[0]);
    const int*   is_numeric = static_cast<const int*>(d_in[1]);
    const int*   emb_idx    = static_cast<const int*>(d_in[2]);
    const float* emb_table  = static_cast<const float*>(d_in[3]);
    const float* W          = static_cast<const float*>(d_in[4]);
    const float* b          = static_cast<const float*>(d_in[5]);
    float*       out        = static_cast<float*>(d_out);

    const int N = in_sizes[0];
    const int rows_per_block = kBlock / 32;          // 8
    const int grid = (N + rows_per_block - 1) / rows_per_block;

    string_numeric_embedding_kernel<<<grid, kBlock, 0, stream>>>(
        values, is_numeric, emb_idx, emb_table, W, b, out, N);
}